// NoiseCorrelation_24111946400298
// MI455X (gfx1250) — compile-verified
//
#include <hip/hip_runtime.h>
#include <stdint.h>

#if __has_builtin(__builtin_amdgcn_wmma_f32_16x16x4_f32)
#define HAVE_WMMA_F32X4 1
#else
#define HAVE_WMMA_F32X4 0
#endif

#if __has_builtin(__builtin_amdgcn_global_load_async_to_lds_b32)
#define HAVE_ASYNC_LDS 1
#else
#define HAVE_ASYNC_LDS 0
#endif

typedef __attribute__((ext_vector_type(2))) float v2f;
typedef __attribute__((ext_vector_type(8))) float v8f;

#define NB   8
#define NC   16
#define HH   192
#define WW   192
#define TY   16      // output tile rows
#define TX   14      // output tile cols
#define CR   18      // corr rows (TY + 2 box halo)
#define CC   16      // corr cols (TX + 2 box halo) == WMMA M
#define IY   26      // input tile rows (CR + 2*4 shift halo)
#define IX   24      // input tile cols (CC + 8 shift halo)
#define DSTRIDE 32   // jj extent per (corr row, corr col): two 16-wide N-chunks

// Fused: correlation (81 shifts, dot over 16 channels) + 3x3 box sum + scale.
// One workgroup = one (b, 16x14) output tile, all 81 planes.
__global__ __launch_bounds__(256)
void noise_corr_fused(const float* __restrict__ noise, float* __restrict__ out)
{
  __shared__ float ls_in[IY * IX * NC];        // 39,936 B  [y][x][c]
  __shared__ float ls_d [CR * CC * DSTRIDE];   // 36,864 B  [r][i][jj]

  const int tid = threadIdx.x;
  const int b   = blockIdx.z;
  const int y0  = blockIdx.y * TY;
  const int x0  = blockIdx.x * TX;

  const float* nbase = noise + (size_t)b * NC * HH * WW;

  // ---------- Stage 0: input tile -> LDS (zero-padded halo) ----------
  // idx order (c,y,x): consecutive lanes hit consecutive gx -> coalesced loads.
  for (int idx = tid; idx < NC * IY * IX; idx += 256) {
    const int c  = idx / (IY * IX);
    const int rr = idx % (IY * IX);
    const int y  = rr / IX;
    const int x  = rr % IX;
    const int gy = y0 - 5 + y;
    const int gx = x0 - 5 + x;
    float* l = &ls_in[(y * IX + x) * NC + c];
    if ((unsigned)gy < HH && (unsigned)gx < WW) {
      const float* g = nbase + (size_t)c * HH * WW + gy * WW + gx;
#if HAVE_ASYNC_LDS
      // CDNA5 async copy to LDS (ASYNCcnt-tracked); per-lane LDS scatter address.
      __builtin_amdgcn_global_load_async_to_lds_b32(
          (__attribute__((address_space(1))) int*)(uintptr_t)g,
          (__attribute__((address_space(3))) int*)(unsigned)(uintptr_t)l,
          0, 0);
#else
      *l = *g;
#endif
    } else {
      *l = 0.0f;   // reproduces both the shift zero-pad and the conv zero-pad
    }
  }
#if HAVE_ASYNC_LDS
#if __has_builtin(__builtin_amdgcn_s_wait_asynccnt)
  __builtin_amdgcn_s_wait_asynccnt(0);
#else
  asm volatile("s_wait_asynccnt 0" ::: "memory");
#endif
#endif
  __syncthreads();

  const int wv    = tid >> 5;    // wave id (8 waves)
  const int lane  = tid & 31;
  const int lhalf = lane >> 4;   // WMMA half-lane split
  const int lmod  = lane & 15;

  // Stage-B fixed mapping: thread -> (y, x) output pixel, pure shift/mask.
  const int ox = tid & 15;       // 0..15, active when < TX
  const int oy = tid >> 4;       // 0..15
  // ls_d element for (dy,dx,dj) relative to (oy,ox):
  //   index = (oy+dy)*CC*DSTRIDE + (ox+dx)*DSTRIDE + (ox+dx) + dj
  //         = oy*512 + ox*33  +  dy*512 + dx*33 + dj   (all-constant tail)
  const float* dbase = &ls_d[oy * (CC * DSTRIDE) + ox * (DSTRIDE + 1)];
  const int    ogx   = x0 + ox;
  const bool   owrite = (ox < TX) & (ogx < WW);
  float* obase = out + (((size_t)b * 81) * HH + (y0 + oy)) * WW + ogx;

#if HAVE_WMMA_F32X4
  // A operand (corr cols x channels) is independent of di: cache in registers.
  // A 16x4 f32 layout: lanes 0-15 M=lane K={0,1}; lanes 16-31 M=lane-16 K={2,3}.
  v2f amat[3][4];
  for (int s = 0; s < 3; ++s) {
    const int r = wv + 8 * s;               // wave-uniform -> EXEC stays full
    if (r < CR) {
      const float* p = &ls_in[((r + 4) * IX + (4 + lmod)) * NC + 2 * lhalf];
      for (int q = 0; q < 4; ++q) amat[s][q] = *(const v2f*)(p + 4 * q);
    }
  }
#endif

  for (int di = 0; di < 9; ++di) {
    // ---------- Stage A: corr band (all dj) for this di -> ls_d ----------
#if HAVE_WMMA_F32X4
    for (int s = 0; s < 3; ++s) {
      const int r = wv + 8 * s;
      if (r < CR) {
        v8f dt[2];
        for (int n = 0; n < 2; ++n) {
          int col = 16 * n + lmod;          // B cols: x' local = jj
          if (col > IX - 1) col = IX - 1;   // n=1,lmod>=8 never consumed; clamp
          const float* p = &ls_in[((r + di) * IX + col) * NC + 2 * lhalf];
          v8f acc = {0.f,0.f,0.f,0.f,0.f,0.f,0.f,0.f};
          for (int q = 0; q < 4; ++q) {     // K=16 channels via 4 chained K=4
            const v2f bq = *(const v2f*)(p + 4 * q);
            acc = __builtin_amdgcn_wmma_f32_16x16x4_f32(
                false, amat[s][q], false, bq, (short)0, acc, false, false);
          }
          dt[n] = acc;
        }
        // D layout: lane j holds column j; VGPR v = row v (lanes<16) / v+8.
        float* drow = &ls_d[r * CC * DSTRIDE + lmod];
        for (int v = 0; v < 8; ++v) {
          const int M = v + 8 * lhalf;
          drow[M * DSTRIDE]      = dt[0][v];   // jj = lmod
          drow[M * DSTRIDE + 16] = dt[1][v];   // jj = 16 + lmod
        }
      }
    }
#else
    // VALU fallback: compute only the needed band values directly.
    for (int idx = tid; idx < CR * CC * 9; idx += 256) {
      const int r  = idx / (CC * 9);
      const int re = idx % (CC * 9);
      const int i  = re / 9;
      const int dj = re % 9;
      const float* pa = &ls_in[((r + 4)  * IX + (4 + i))  * NC];
      const float* pb = &ls_in[((r + di) * IX + (i + dj)) * NC];
      float sum = 0.f;
      for (int c = 0; c < NC; ++c) sum += pa[c] * pb[c];
      ls_d[(r * CC + i) * DSTRIDE + (i + dj)] = sum;
    }
#endif
    __syncthreads();

    // ---------- Stage B: 3x3 box sum + scale + non-temporal store ----------
    // Fully unrolled; all 81 LDS reads use immediate offsets off one base,
    // 9 NT stores use immediate plane offsets off one base.
    if (owrite) {
      const float* dj0 = dbase;
      float* op = obase + (size_t)(di * 9) * HH * WW;
#pragma unroll
      for (int dj = 0; dj < 9; ++dj) {
        float sum = 0.f;
#pragma unroll
        for (int dy = 0; dy < 3; ++dy)
#pragma unroll
          for (int dx = 0; dx < 3; ++dx)
            sum += dj0[dy * (CC * DSTRIDE) + dx * (DSTRIDE + 1) + dj];
        __builtin_nontemporal_store(sum * (1.0f / 144.0f),
                                    op + (size_t)dj * HH * WW);
      }
    }
    __syncthreads();
  }
}

extern "C" void kernel_launch(void* const* d_in, const int* in_sizes, int n_in,
                              void* d_out, int out_size, void* d_ws, size_t ws_size,
                              hipStream_t stream) {
  (void)in_sizes; (void)n_in; (void)d_ws; (void)ws_size; (void)out_size;
  const float* noise = (const float*)d_in[0];
  float* out = (float*)d_out;
  dim3 grid((WW + TX - 1) / TX, HH / TY, NB);   // 14 x 12 x 8 = 1344 blocks
  noise_corr_fused<<<grid, 256, 0, stream>>>(noise, out);
}